// GATConv_46840913330824
// MI455X (gfx1250) — compile-verified
//
#include <hip/hip_runtime.h>
#include <hip/hip_bf16.h>
#include <math.h>

// ---------------------------------------------------------------------------
// 2-layer GAT for MI455X (gfx1250, wave32).
//   layer(x): h = x @ W_src (WMMA f32 16x16x4)
//             a_src[n] = h[n] . att_src
//             a_dst[n] = x[n] . (W_dst @ att_dst)        <- avoids 2nd GEMM
//             e = leaky_relu(a_src[src] + a_dst[dst])
//             segment-softmax over dst (atomicMax via ordered-uint, atomicAdd)
//             out[dst] += alpha * h[src]  (coalesced float4 gather + f32 atomics)
//             out += bias (+ReLU for layer 1)
// ---------------------------------------------------------------------------

#define KDIM 256          // inner dim is 256 for both layers
#define NEG_SLOPE 0.2f
#define ENC_NEG_INF 0x007FFFFFu   // enc(-inf)

typedef float v2f __attribute__((ext_vector_type(2)));
typedef float v8f __attribute__((ext_vector_type(8)));

// Order-preserving f32 <-> u32 mapping so atomicMax(u32) == float max.
__device__ __forceinline__ unsigned enc_f32(float f) {
  unsigned u = __float_as_uint(f);
  return (u & 0x80000000u) ? ~u : (u | 0x80000000u);
}
__device__ __forceinline__ float dec_f32(unsigned u) {
  return (u & 0x80000000u) ? __uint_as_float(u & 0x7FFFFFFFu)
                           : __uint_as_float(~u);
}

// ---------------------------------------------------------------------------
// H[N,C] = X[N,256] @ W[256,C].  grid = (ceil(N/16), C/128), block = 256 (8 waves).
// X strip staged in LDS (stride 260 -> 16 rows hit distinct banks for fixed k).
// Each wave: one 16x16 output tile, 64 chained v_wmma_f32_16x16x4_f32.
// C is a template parameter so W addressing strength-reduces and the epilogue
// becomes 8 immediate-offset stores off one base pointer (no per-row branches).
// ---------------------------------------------------------------------------
template <int C>
__global__ __launch_bounds__(256) void gemm_xw_wmma(
    const float* __restrict__ X, const float* __restrict__ W,
    float* __restrict__ H, int N) {
  __shared__ float sx[16][KDIM + 4];
  const int row0 = blockIdx.x * 16;
  const int colBase = blockIdx.y * 128;
  const int tid = threadIdx.x;
  const bool full = (row0 + 16 <= N);

  // Cooperative load: 16 rows x 256 cols = 1024 float4, 4 per thread.
  if (full) {
#pragma unroll
    for (int i = 0; i < 4; ++i) {
      int idx = tid + i * 256;          // float4 index 0..1023
      int r = idx >> 6;                 // 64 float4 per row
      int c4 = (idx & 63) << 2;
      float4 v = *(const float4*)(X + (size_t)(row0 + r) * KDIM + c4);
      sx[r][c4 + 0] = v.x; sx[r][c4 + 1] = v.y;
      sx[r][c4 + 2] = v.z; sx[r][c4 + 3] = v.w;
    }
  } else {
#pragma unroll
    for (int i = 0; i < 4; ++i) {
      int idx = tid + i * 256;
      int r = idx >> 6;
      int c4 = (idx & 63) << 2;
      float4 v = make_float4(0.f, 0.f, 0.f, 0.f);
      if (row0 + r < N) v = *(const float4*)(X + (size_t)(row0 + r) * KDIM + c4);
      sx[r][c4 + 0] = v.x; sx[r][c4 + 1] = v.y;
      sx[r][c4 + 2] = v.z; sx[r][c4 + 3] = v.w;
    }
  }
  __syncthreads();

  const int wave = tid >> 5;
  const int lane = tid & 31;
  const int m  = lane & 15;           // M (A) / N (B) index
  const int hi = lane >> 4;           // K half select
  const int col = colBase + wave * 16 + m;

  const float* wp = W + col;          // column base; rows at constant stride C
  v8f acc = {};
#pragma unroll 4
  for (int k = 0; k < KDIM; k += 4) {
    const int kk = k + hi * 2;
    // A 16x4 f32 (ISA layout): lanes 0-15 hold K0,K1; lanes 16-31 hold K2,K3
    v2f a; a.x = sx[m][kk]; a.y = sx[m][kk + 1];
    // B 4x16 f32: lane = column, same K-half split as A
    v2f b; b.x = wp[(size_t)kk * C]; b.y = wp[(size_t)(kk + 1) * C];
    acc = __builtin_amdgcn_wmma_f32_16x16x4_f32(
        false, a, false, b, (short)0, acc, false, false);
  }

  // C/D layout: VGPR r -> row r (lanes 0-15) / row r+8 (lanes 16-31)
  float* hp = H + (size_t)(row0 + hi * 8) * C + col;
  if (full) {
#pragma unroll
    for (int r = 0; r < 8; ++r) hp[(size_t)r * C] = acc[r];
  } else {
#pragma unroll
    for (int r = 0; r < 8; ++r)
      if (row0 + hi * 8 + r < N) hp[(size_t)r * C] = acc[r];
  }
}

// v[k] = sum_c W[k,c] * a[c]   (tiny: K threads)
__global__ void matvec_rows(const float* __restrict__ W,
                            const float* __restrict__ a,
                            float* __restrict__ v, int K, int C) {
  int k = blockIdx.x * blockDim.x + threadIdx.x;
  if (k >= K) return;
  float s = 0.f;
  for (int c = 0; c < C; ++c) s += W[(size_t)k * C + c] * a[c];
  v[k] = s;
}

// Per-node attention logits: one wave per node, shuffle reduction (wave32).
__global__ __launch_bounds__(256) void node_att(
    const float* __restrict__ Hs, const float* __restrict__ att_s,
    const float* __restrict__ X, const float* __restrict__ v,
    float* __restrict__ a_src, float* __restrict__ a_dst, int N, int C) {
  int node = blockIdx.x * 8 + (threadIdx.x >> 5);
  int lane = threadIdx.x & 31;
  if (node >= N) return;
  float s1 = 0.f, s2 = 0.f;
  for (int c = lane; c < C; c += 32) s1 += Hs[(size_t)node * C + c] * att_s[c];
  for (int k = lane; k < KDIM; k += 32) s2 += X[(size_t)node * KDIM + k] * v[k];
#pragma unroll
  for (int o = 16; o > 0; o >>= 1) {
    s1 += __shfl_xor(s1, o, 32);
    s2 += __shfl_xor(s2, o, 32);
  }
  if (lane == 0) { a_src[node] = s1; a_dst[node] = s2; }
}

__global__ void init_nodes(unsigned* __restrict__ m_enc,
                           float* __restrict__ s, int N) {
  int i = blockIdx.x * blockDim.x + threadIdx.x;
  if (i < N) { m_enc[i] = ENC_NEG_INF; s[i] = 0.f; }
}

__global__ void zero_f32(float* __restrict__ p, size_t n) {
  size_t i = (size_t)blockIdx.x * blockDim.x + threadIdx.x;
  if (i < n) p[i] = 0.f;
}

// e = leaky_relu(a_src[src] + a_dst[dst]); segment max into m_enc[dst]
__global__ void edge_logits(const int* __restrict__ src,
                            const int* __restrict__ dst,
                            const float* __restrict__ a_src,
                            const float* __restrict__ a_dst,
                            float* __restrict__ ebuf,
                            unsigned* __restrict__ m_enc, int E) {
  int e = blockIdx.x * blockDim.x + threadIdx.x;
  if (e >= E) return;
  int sn = src[e], dn = dst[e];
  float lg = a_src[sn] + a_dst[dn];
  lg = (lg > 0.f) ? lg : NEG_SLOPE * lg;
  ebuf[e] = lg;
  atomicMax(&m_enc[dn], enc_f32(lg));
}

// decode segment max; empty segment (-inf) -> 0, matching reference isfinite()
__global__ void m_finalize(float* __restrict__ mbuf, int N) {
  int i = blockIdx.x * blockDim.x + threadIdx.x;
  if (i >= N) return;
  float m = dec_f32(((const unsigned*)mbuf)[i]);
  if (!isfinite(m)) m = 0.f;
  mbuf[i] = m;
}

// ex = exp(e - m[dst]); segment sum into s[dst]
__global__ void edge_exp(const int* __restrict__ dst,
                         float* __restrict__ ebuf,
                         const float* __restrict__ m,
                         float* __restrict__ s, int E) {
  int e = blockIdx.x * blockDim.x + threadIdx.x;
  if (e >= E) return;
  int dn = dst[e];
  float ex = expf(ebuf[e] - m[dn]);
  ebuf[e] = ex;
  atomicAdd(&s[dn], ex);
}

// out[dst] += alpha * h[src]; one wave per edge, float4 gather, f32 atomics.
// h and out (~51 MB each) are L2-resident on MI455X's 192 MB L2.
// C templated so the per-lane loop fully unrolls (2 iters @256, 1 @128).
template <int C>
__global__ __launch_bounds__(256) void edge_aggregate(
    const int* __restrict__ src, const int* __restrict__ dst,
    const float* __restrict__ ebuf, const float* __restrict__ s,
    const float* __restrict__ Hs, float* __restrict__ out, int E) {
  int e = blockIdx.x * 8 + (threadIdx.x >> 5);
  int lane = threadIdx.x & 31;
  if (e >= E) return;
  int sn = src[e], dn = dst[e];
  float alpha = ebuf[e] / (s[dn] + 1e-16f);
  const float4* hp = (const float4*)(Hs + (size_t)sn * C);
  float* op = out + (size_t)dn * C;
#pragma unroll
  for (int j = lane; j < (C >> 2); j += 32) {
    float4 hv = hp[j];
    int c = j << 2;
    atomicAdd(op + c + 0, alpha * hv.x);
    atomicAdd(op + c + 1, alpha * hv.y);
    atomicAdd(op + c + 2, alpha * hv.z);
    atomicAdd(op + c + 3, alpha * hv.w);
  }
}

__global__ void bias_act(float* __restrict__ H, const float* __restrict__ b,
                         size_t total, int Cmask, int do_relu) {
  size_t i = (size_t)blockIdx.x * blockDim.x + threadIdx.x;
  if (i >= total) return;
  float v = H[i] + b[i & (size_t)Cmask];
  if (do_relu) v = v > 0.f ? v : 0.f;
  H[i] = v;
}

// ---------------------------------------------------------------------------
template <int C>
static void run_layer(const float* X, const float* Wsrc, const float* Wdst,
                      const float* att_s, const float* att_d, const float* bias,
                      const int* src, const int* dst,
                      float* Hs, float* out,
                      float* ebuf, float* asrc, float* adst,
                      float* mbuf, float* sbuf, float* vbuf,
                      int N, int E, int do_relu, hipStream_t stream) {
  const int nB = 256;
  dim3 ggrid((N + 15) / 16, C / 128);
  gemm_xw_wmma<C><<<ggrid, nB, 0, stream>>>(X, Wsrc, Hs, N);
  matvec_rows<<<1, KDIM, 0, stream>>>(Wdst, att_d, vbuf, KDIM, C);
  node_att<<<(N + 7) / 8, nB, 0, stream>>>(Hs, att_s, X, vbuf, asrc, adst, N, C);
  init_nodes<<<(N + nB - 1) / nB, nB, 0, stream>>>((unsigned*)mbuf, sbuf, N);
  size_t total = (size_t)N * C;
  zero_f32<<<(int)((total + nB - 1) / nB), nB, 0, stream>>>(out, total);
  edge_logits<<<(E + nB - 1) / nB, nB, 0, stream>>>(src, dst, asrc, adst, ebuf,
                                                    (unsigned*)mbuf, E);
  m_finalize<<<(N + nB - 1) / nB, nB, 0, stream>>>(mbuf, N);
  edge_exp<<<(E + nB - 1) / nB, nB, 0, stream>>>(dst, ebuf, mbuf, sbuf, E);
  edge_aggregate<C><<<(E + 7) / 8, nB, 0, stream>>>(src, dst, ebuf, sbuf, Hs,
                                                    out, E);
  bias_act<<<(int)((total + nB - 1) / nB), nB, 0, stream>>>(out, bias, total,
                                                            C - 1, do_relu);
}

extern "C" void kernel_launch(void* const* d_in, const int* in_sizes, int n_in,
                              void* d_out, int out_size, void* d_ws,
                              size_t ws_size, hipStream_t stream) {
  const float* x      = (const float*)d_in[0];
  const int*   edge   = (const int*)d_in[1];
  const float* W1s    = (const float*)d_in[2];
  const float* W1d    = (const float*)d_in[3];
  const float* att1s  = (const float*)d_in[4];
  const float* att1d  = (const float*)d_in[5];
  const float* b1     = (const float*)d_in[6];
  const float* W2s    = (const float*)d_in[7];
  const float* W2d    = (const float*)d_in[8];
  const float* att2s  = (const float*)d_in[9];
  const float* att2d  = (const float*)d_in[10];
  const float* b2     = (const float*)d_in[11];

  const int N = in_sizes[0] / KDIM;   // 50000
  const int E = in_sizes[1] / 2;      // 800000
  const int HID = 256;
  const int* src = edge;
  const int* dst = edge + E;

  // Workspace layout (floats): ~107 MB total
  float* ws   = (float*)d_ws;
  float* hA   = ws;                        // [N,256] projection h = x@W_src
  float* hB   = hA + (size_t)N * HID;      // [N,256] layer1 out / layer2 in
  float* ebuf = hB + (size_t)N * HID;      // [E] logits -> exp
  float* asrc = ebuf + E;                  // [N]
  float* adst = asrc + N;                  // [N]
  float* mbuf = adst + N;                  // [N] enc-max -> max
  float* sbuf = mbuf + N;                  // [N] softmax denom
  float* vbuf = sbuf + N;                  // [256] W_dst @ att_dst

  // Layer 1: 256 -> 256, ReLU
  run_layer<256>(x, W1s, W1d, att1s, att1d, b1, src, dst, hA, hB,
                 ebuf, asrc, adst, mbuf, sbuf, vbuf, N, E, 1, stream);
  // Layer 2: 256 -> 128, no activation, writes d_out
  run_layer<128>(hB, W2s, W2d, att2s, att2d, b2, src, dst, hA, (float*)d_out,
                 ebuf, asrc, adst, mbuf, sbuf, vbuf, N, E, 0, stream);
}